// SDFChamferLoss_60189671686288
// MI455X (gfx1250) — compile-verified
//
#include <hip/hip_runtime.h>
#include <math.h>

// ---------------------------------------------------------------------------
// SDF Chamfer pseudo-loss for MI455X (gfx1250).
// Heavy stage: 8064x8064 pairwise distances per batch, mapped onto
// V_WMMA_F32_16X16X4_F32: A row = (pr, pc, pr^2+pc^2, 1),
//                         B col = (-2gr, -2gc, 1, gr^2+gc^2 [+1e20 if masked])
//  =>  A.B = d^2 (masked columns land at ~1e20, sqrt -> BIG=1e10).
// Inner loop tracks d^2 (sqrt is monotone; applied once per row at the end).
// ---------------------------------------------------------------------------

typedef __attribute__((ext_vector_type(2))) float v2f;
typedef __attribute__((ext_vector_type(8))) float v8f;

#define SDF_EPS   1e-3f
#define SDF_BAND  3.0f
#define SDF_MASK2 1.0e20f      // masked-column d^2 (sqrt -> 1e10 == BIG)
#define SDF_INIT  3.0e38f

// ---------------- Stage 1: zero-crossing point extraction ------------------
__global__ void sdf_points_kernel(const float* __restrict__ sdf,
                                  float* __restrict__ Apack,   // [B][N][4]
                                  float* __restrict__ Bpack,   // [B][N][4]
                                  float* __restrict__ pos,     // [B][N][2]
                                  float* __restrict__ valid,   // [B][N]
                                  int Bn, int H, int W, int N)
{
    int tid = blockIdx.x * blockDim.x + threadIdx.x;
    int total = Bn * N;
    if (tid >= total) return;
    int b = tid / N, n = tid % N;
    const float* s = sdf + (size_t)b * H * W;
    const int NV = (H - 1) * W;

    float v1, v2, pr, pc;
    if (n < NV) {                       // vertical crossings
        int r0 = n / W, c = n % W;
        v1 = s[r0 * W + c];
        v2 = s[(r0 + 1) * W + c];
        float a = fminf(fmaxf(fabsf(v1) / fmaxf(fabsf(v1) + fabsf(v2), 1e-8f), 0.0f), 1.0f);
        pr = (float)r0 + a; pc = (float)c;
    } else {                            // horizontal crossings
        int j = n - NV;
        int r = j / (W - 1), c0 = j % (W - 1);
        v1 = s[r * W + c0];
        v2 = s[r * W + c0 + 1];
        float a = fminf(fmaxf(fabsf(v1) / fmaxf(fabsf(v1) + fabsf(v2), 1e-8f), 0.0f), 1.0f);
        pr = (float)r; pc = (float)c0 + a;
    }
    bool vld = ((v1 < 0.f && v2 > 0.f) || (v1 > 0.f && v2 < 0.f) ||
                (fabsf(v1) <= SDF_EPS) || (fabsf(v2) <= SDF_EPS));

    size_t o = (size_t)b * N + n;
    float r2 = pr * pr + pc * pc;
    Apack[o * 4 + 0] = pr;
    Apack[o * 4 + 1] = pc;
    Apack[o * 4 + 2] = r2;
    Apack[o * 4 + 3] = 1.0f;
    Bpack[o * 4 + 0] = -2.0f * pr;
    Bpack[o * 4 + 1] = -2.0f * pc;
    Bpack[o * 4 + 2] = 1.0f;
    Bpack[o * 4 + 3] = r2 + (vld ? 0.0f : SDF_MASK2);   // column mask folded in
    pos[o * 2 + 0] = pr;
    pos[o * 2 + 1] = pc;
    valid[o] = vld ? 1.0f : 0.0f;
}

// ---------------- Stage 2: WMMA distance strips -----------------------------
// One full wave per 16-row P strip sweeps 504 G tiles. Per tile:
//   b64 load of B operand -> one V_WMMA_F32_16X16X4_F32 -> running d^2 argmin.
// D layout (ISA 7.12.2): VGPR k -> row k + 8*(lane>=16), col = lane&15.
__global__ void __launch_bounds__(32)
sdf_strip_kernel(const float* __restrict__ Apack,   // pred points, A form
                 const float* __restrict__ Bpack,   // gt points, B form (masked)
                 float* __restrict__ rowMin,        // [B][N] (distance, post-sqrt)
                 unsigned* __restrict__ rowArg,     // [B][N]
                 int Npts)
{
    const int numTiles = Npts >> 4;              // 504
    const int strip = blockIdx.x;                // b*numTiles + pt
    const int b  = strip / numTiles;
    const int pt = strip % numTiles;
    const int lane = threadIdx.x;                // 0..31 (wave32)
    const int half = lane >> 4;                  // 0 or 1
    const int l16  = lane & 15;
    const int k0   = half * 2;                   // K components {0,1} or {2,3}

    const float* Ab = Apack + (size_t)b * Npts * 4;
    const float* Bb = Bpack + (size_t)b * Npts * 4;

    // A operand (fixed for the strip): lane = M row, VGPR0/1 = K=k0,k0+1
    const int prow = pt * 16 + l16;
    float2 av = *(const float2*)(Ab + (size_t)prow * 4 + k0);
    v2f a; a.x = av.x; a.y = av.y;

    float    rmin[8];                            // running min of d^2
    unsigned rarg[8];
#pragma unroll
    for (int k = 0; k < 8; ++k) { rmin[k] = SDF_INIT; rarg[k] = 0u; }

#pragma unroll 4
    for (int t = 0; t < numTiles; ++t) {
        const int g = t * 16 + l16;
        float2 bv = *(const float2*)(Bb + (size_t)g * 4 + k0);
        v2f bm; bm.x = bv.x; bm.y = bv.y;

        // prefetch a few tiles ahead (global_prefetch path; uniform branch-free)
        int tp = min(t + 8, numTiles - 1);
        __builtin_prefetch(Bb + (size_t)(tp * 16 + l16) * 4 + k0, 0, 3);

        v8f c = {};
        c = __builtin_amdgcn_wmma_f32_16x16x4_f32(
                /*neg_a=*/false, a, /*neg_b=*/false, bm,
                /*c_mod=*/(short)0, c, /*reuse_a=*/false, /*reuse_b=*/false);

#pragma unroll
        for (int k = 0; k < 8; ++k) {
            float d2 = c[k];                      // d^2 (monotone proxy for d)
            if (d2 < rmin[k]) { rmin[k] = d2; rarg[k] = (unsigned)g; }
        }
    }

    // Butterfly (min,argmin) over the 16 lanes of each half-wave.
#pragma unroll
    for (int k = 0; k < 8; ++k) {
        float v = rmin[k]; unsigned i = rarg[k];
#pragma unroll
        for (int m = 1; m <= 8; m <<= 1) {
            float    ov = __shfl_xor(v, m, 32);
            unsigned oi = (unsigned)__shfl_xor((int)i, m, 32);
            if (ov < v || (ov == v && oi < i)) { v = ov; i = oi; }
        }
        if (l16 == 0) {                           // lanes 0 and 16 publish
            int row = pt * 16 + half * 8 + k;
            rowMin[(size_t)b * Npts + row] = sqrtf(fmaxf(v, 0.0f));  // back to d
            rowArg[(size_t)b * Npts + row] = i;
        }
    }
}

// ---------------- Stage 3: pseudo-loss per batch ----------------------------
__device__ inline float grad_r(const float* s, int r, int c, int H, int W) {
    if (r == 0)      return s[W + c] - s[c];
    if (r == H - 1)  return s[(H - 1) * W + c] - s[(H - 2) * W + c];
    return 0.5f * (s[(r + 1) * W + c] - s[(r - 1) * W + c]);
}
__device__ inline float grad_c(const float* s, int r, int c, int H, int W) {
    if (c == 0)      return s[r * W + 1] - s[r * W];
    if (c == W - 1)  return s[r * W + W - 1] - s[r * W + W - 2];
    return 0.5f * (s[r * W + c + 1] - s[r * W + c - 1]);
}

// splat-then-dot(pred) == proj * bilinear(pred, P), so grad_map is never built.
__global__ void sdf_finalize_kernel(const float* __restrict__ pred,
                                    const float* __restrict__ Ppos,
                                    const float* __restrict__ vP,
                                    const float* __restrict__ Gpos,
                                    const float* __restrict__ rowMin,
                                    const unsigned* __restrict__ rowArg,
                                    float* __restrict__ pseudo,   // [B]
                                    int N, int H, int W)
{
    const int b = blockIdx.x;
    const float* pb = pred + (size_t)b * H * W;
    const float* Pp = Ppos + (size_t)b * N * 2;
    const float* Gp = Gpos + (size_t)b * N * 2;
    const float* vp = vP + (size_t)b * N;
    const float* rm = rowMin + (size_t)b * N;
    const unsigned* ra = rowArg + (size_t)b * N;

    float l1n = 0.0f, contrib = 0.0f;
    for (int n = threadIdx.x; n < N; n += blockDim.x) {
        if (vp[n] == 0.0f) continue;
        if (rm[n] > SDF_BAND) continue;
        float pr = Pp[n * 2 + 0], pc = Pp[n * 2 + 1];
        unsigned idx = ra[n];
        float dlr = Gp[(size_t)idx * 2 + 0] - pr;
        float dlc = Gp[(size_t)idx * 2 + 1] - pc;

        int r0 = min(max((int)floorf(pr), 0), H - 1);
        int c0 = min(max((int)floorf(pc), 0), W - 1);
        int r1 = min(r0 + 1, H - 1);
        int c1 = min(c0 + 1, W - 1);
        float dr = pr - (float)r0, dc = pc - (float)c0;
        float w00 = (1.f - dr) * (1.f - dc), w01 = (1.f - dr) * dc;
        float w10 = dr * (1.f - dc),         w11 = dr * dc;

        float nr = w00 * grad_r(pb, r0, c0, H, W) + w01 * grad_r(pb, r0, c1, H, W)
                 + w10 * grad_r(pb, r1, c0, H, W) + w11 * grad_r(pb, r1, c1, H, W);
        float nc = w00 * grad_c(pb, r0, c0, H, W) + w01 * grad_c(pb, r0, c1, H, W)
                 + w10 * grad_c(pb, r1, c0, H, W) + w11 * grad_c(pb, r1, c1, H, W);

        l1n += fabsf(nr) + fabsf(nc);

        float proj = dlr * nr + dlc * nc;         // SCALE == 1
        float sp = w00 * pb[r0 * W + c0] + w01 * pb[r0 * W + c1]
                 + w10 * pb[r1 * W + c0] + w11 * pb[r1 * W + c1];
        contrib += proj * sp;
    }

    __shared__ float sA[256], sB[256];
    sA[threadIdx.x] = l1n; sB[threadIdx.x] = contrib;
    __syncthreads();
    for (int s = 128; s > 0; s >>= 1) {
        if ((int)threadIdx.x < s) { sA[threadIdx.x] += sA[threadIdx.x + s];
                                    sB[threadIdx.x] += sB[threadIdx.x + s]; }
        __syncthreads();
    }
    if (threadIdx.x == 0)
        pseudo[b] = -sB[0] / (sA[0] + 1e-8f);     // deferred NORM_N division
}

// ---------------- L1 term + final scalar ------------------------------------
__global__ void sdf_zero_kernel(float* acc) { if (threadIdx.x == 0) acc[0] = 0.0f; }

__global__ void sdf_l1_kernel(const float* __restrict__ a, const float* __restrict__ b,
                              float* __restrict__ acc, int n)
{
    __shared__ float sm[256];
    float s = 0.0f;
    for (int i = blockIdx.x * blockDim.x + threadIdx.x; i < n; i += gridDim.x * blockDim.x)
        s += fabsf(a[i] - b[i]);
    sm[threadIdx.x] = s;
    __syncthreads();
    for (int k = 128; k > 0; k >>= 1) {
        if ((int)threadIdx.x < k) sm[threadIdx.x] += sm[threadIdx.x + k];
        __syncthreads();
    }
    if (threadIdx.x == 0) atomicAdd(acc, sm[0]);
}

__global__ void sdf_loss_kernel(const float* __restrict__ pseudo,
                                const float* __restrict__ l1sum,
                                float* __restrict__ out, int Bn, int total)
{
    float s = 0.0f;
    for (int b = 0; b < Bn; ++b) s += pseudo[b];
    out[0] = s / (float)Bn + l1sum[0] / (float)total;
}

// ---------------------------------------------------------------------------
extern "C" void kernel_launch(void* const* d_in, const int* in_sizes, int n_in,
                              void* d_out, int out_size, void* d_ws, size_t ws_size,
                              hipStream_t stream)
{
    const float* pred = (const float*)d_in[0];
    const float* gt   = (const float*)d_in[1];
    const int H = 64, W = 64;
    const int Bn = in_sizes[0] / (H * W);              // 4
    const int N  = (H - 1) * W + H * (W - 1);          // 8064 = 16*504
    const int BN = Bn * N;

    // workspace carve-up (floats)
    float* w = (float*)d_ws;
    size_t off = 0;
    float* PA   = w + off; off += (size_t)BN * 4;      // pred, A form
    float* PB   = w + off; off += (size_t)BN * 4;      // pred, B form (unused)
    float* GA   = w + off; off += (size_t)BN * 4;      // gt, A form (unused)
    float* GB   = w + off; off += (size_t)BN * 4;      // gt, B form (mask folded)
    float* Ppos = w + off; off += (size_t)BN * 2;
    float* Gpos = w + off; off += (size_t)BN * 2;
    float* vP   = w + off; off += (size_t)BN;
    float* vG   = w + off; off += (size_t)BN;          // written, unused downstream
    float* rowMin = w + off; off += (size_t)BN;
    unsigned* rowArg = (unsigned*)(w + off); off += (size_t)BN;
    float* pseudo = w + off; off += (size_t)Bn;
    float* l1sum  = w + off; off += 1;

    // Stage 1: points for pred and gt
    {
        int total = BN;
        int blocks = (total + 255) / 256;
        sdf_points_kernel<<<blocks, 256, 0, stream>>>(pred, PA, PB, Ppos, vP, Bn, H, W, N);
        sdf_points_kernel<<<blocks, 256, 0, stream>>>(gt,   GA, GB, Gpos, vG, Bn, H, W, N);
    }

    // Stage 2: WMMA distance strips (row-min + argmin over G; mask pre-folded)
    {
        int strips = Bn * (N / 16);                    // 2016 waves
        sdf_strip_kernel<<<strips, 32, 0, stream>>>(PA, GB, rowMin, rowArg, N);
    }

    // Stage 3: L1 term + per-batch pseudo + final scalar
    sdf_zero_kernel<<<1, 32, 0, stream>>>(l1sum);
    sdf_l1_kernel<<<64, 256, 0, stream>>>(pred, gt, l1sum, Bn * H * W);
    sdf_finalize_kernel<<<Bn, 256, 0, stream>>>(pred, Ppos, vP, Gpos, rowMin, rowArg,
                                                pseudo, N, H, W);
    sdf_loss_kernel<<<1, 1, 0, stream>>>(pseudo, l1sum, (float*)d_out, Bn, Bn * H * W);
}